// OriginalSparseGLU_7327214207124
// MI455X (gfx1250) — compile-verified
//
#include <hip/hip_runtime.h>
#include <cmath>

typedef __attribute__((ext_vector_type(2))) float v2f;
typedef __attribute__((ext_vector_type(4))) float f4;
typedef __attribute__((ext_vector_type(8))) float v8f;

#define NEXP 64
#define TPE  64
#define HDIM 1024
#define FDIM 1024
#define KC   32
#define AST  36   // padded LDS row stride (floats) for 32-wide K tiles
#define BST  68   // padded LDS row stride (floats) for 64-wide N tiles

#define USE_ASYNC_LDS 1

__device__ __forceinline__ float gelu_exact(float x) {
    return 0.5f * x * (1.0f + erff(x * 0.70710678118654752f));
}

// LDS byte offset of a generic pointer into __shared__ (addrspacecast to AS3).
__device__ __forceinline__ unsigned lds_off(const void* p) {
    return (unsigned)(uintptr_t)(__attribute__((address_space(3))) const void*)p;
}

#if USE_ASYNC_LDS
// GLOBAL_LOAD_ASYNC_TO_LDS_B128: DMA 16B cache->LDS, tracked by ASYNCcnt.
__device__ __forceinline__ void async_copy_b128(unsigned lds, const void* gptr) {
    asm volatile("global_load_async_to_lds_b128 %0, %1, off"
                 :: "v"(lds), "v"(gptr) : "memory");
}
__device__ __forceinline__ void wait_async0() {
    asm volatile("s_wait_asynccnt 0" ::: "memory");
}
#else
__device__ __forceinline__ void async_copy_b128(unsigned lds, const void* gptr) {
    // sync fallback: VGPR round-trip
    *(f4*)((__attribute__((address_space(3))) float*)(uintptr_t)lds) = *(const f4*)gptr;
}
__device__ __forceinline__ void wait_async0() {}
#endif

// ---------------------------------------------------------------------------
// Pass 1: per (expert, f-block): h1 = x@w1^T, h2 = x@v1^T, g = gelu(h1)*h2
// block = 128 threads (4 waves); grid = NEXP * (FDIM/64)
// ---------------------------------------------------------------------------
__global__ __launch_bounds__(128, 2)
void moe_glu_pass1(const float* __restrict__ x,
                   const float* __restrict__ w1,
                   const float* __restrict__ v1,
                   float* __restrict__ g)
{
    __shared__ float xs[2][TPE * AST];
    __shared__ float ws[2][64 * AST];
    __shared__ float vs[2][64 * AST];

    const int e    = blockIdx.x >> 4;          // FDIM/64 == 16 f-blocks
    const int fblk = (blockIdx.x & 15) * 64;
    const int tid  = threadIdx.x;
    const int wave = tid >> 5;                 // 0..3 -> N subtile
    const int lane = tid & 31;

    const float* xe = x  + (size_t)e * TPE  * HDIM;
    const float* we = w1 + (size_t)e * FDIM * HDIM + (size_t)fblk * HDIM;
    const float* ve = v1 + (size_t)e * FDIM * HDIM + (size_t)fblk * HDIM;

    // per-thread slice of the cooperative tile copy (4 b128 per array)
    const int rowc[4] = { (tid + 0*128) >> 3, (tid + 1*128) >> 3,
                          (tid + 2*128) >> 3, (tid + 3*128) >> 3 };
    const int colc    = (tid & 7) * 4;

    const v8f vzero = {0.f,0.f,0.f,0.f,0.f,0.f,0.f,0.f};
    v8f acc1[4], acc2[4];
#pragma unroll
    for (int i = 0; i < 4; ++i) { acc1[i] = vzero; acc2[i] = vzero; }

    const int mrow = lane & 15;
    const int koff = (lane >> 4) << 1;   // K pair select: 0 / 2
    const int ncol = lane & 15;

    // issue tile 0
#pragma unroll
    for (int i = 0; i < 4; ++i) {
        size_t go = (size_t)rowc[i] * HDIM + colc;
        int    lo = rowc[i] * AST + colc;
        async_copy_b128(lds_off(&xs[0][lo]), xe + go);
        async_copy_b128(lds_off(&ws[0][lo]), we + go);
        async_copy_b128(lds_off(&vs[0][lo]), ve + go);
    }

    const int nTiles = HDIM / KC;
    for (int t = 0; t < nTiles; ++t) {
        wait_async0();
        __syncthreads();                 // buffer t%2 fully populated, all waves past t-1

        if (t + 1 < nTiles) {            // prefetch tile t+1 into the other buffer
            const int b = (t + 1) & 1;
            const int k0 = (t + 1) * KC;
#pragma unroll
            for (int i = 0; i < 4; ++i) {
                size_t go = (size_t)rowc[i] * HDIM + k0 + colc;
                int    lo = rowc[i] * AST + colc;
                async_copy_b128(lds_off(&xs[b][lo]), xe + go);
                async_copy_b128(lds_off(&ws[b][lo]), we + go);
                async_copy_b128(lds_off(&vs[b][lo]), ve + go);
            }
        }

        const float* xb = xs[t & 1];
        const float* wb = ws[t & 1];
        const float* vb = vs[t & 1];

#pragma unroll
        for (int kk = 0; kk < KC; kk += 4) {
            const int kb = kk + koff;
            v2f b1, b2;
            b1.x = wb[(wave * 16 + ncol) * AST + kb];
            b1.y = wb[(wave * 16 + ncol) * AST + kb + 1];
            b2.x = vb[(wave * 16 + ncol) * AST + kb];
            b2.y = vb[(wave * 16 + ncol) * AST + kb + 1];
#pragma unroll
            for (int mi = 0; mi < 4; ++mi) {
                v2f a;
                a.x = xb[(mi * 16 + mrow) * AST + kb];
                a.y = xb[(mi * 16 + mrow) * AST + kb + 1];
                acc1[mi] = __builtin_amdgcn_wmma_f32_16x16x4_f32(
                    false, a, false, b1, (short)0, acc1[mi], false, false);
                acc2[mi] = __builtin_amdgcn_wmma_f32_16x16x4_f32(
                    false, a, false, b2, (short)0, acc2[mi], false, false);
            }
        }
    }

    // epilogue: g = gelu(h1) * h2   (C/D layout: N=lane&15, M=vgpr + 8*lane[4])
    float* ge = g + (size_t)e * TPE * FDIM;
    const int nOut = fblk + wave * 16 + (lane & 15);
    const int mhi  = (lane >> 4) * 8;
#pragma unroll
    for (int mi = 0; mi < 4; ++mi) {
#pragma unroll
        for (int r = 0; r < 8; ++r) {
            int tt = mi * 16 + mhi + r;
            ge[(size_t)tt * FDIM + nOut] = gelu_exact(acc1[mi][r]) * acc2[mi][r];
        }
    }
}

// ---------------------------------------------------------------------------
// Pass 2: per (expert, h-block): out = g @ w2
// ---------------------------------------------------------------------------
__global__ __launch_bounds__(128, 2)
void moe_glu_pass2(const float* __restrict__ g,
                   const float* __restrict__ w2,
                   float* __restrict__ out)
{
    __shared__ float gs[2][TPE * AST];
    __shared__ float bs[2][KC * BST];

    const int e    = blockIdx.x >> 4;          // HDIM/64 == 16 h-blocks
    const int hblk = (blockIdx.x & 15) * 64;
    const int tid  = threadIdx.x;
    const int wave = tid >> 5;
    const int lane = tid & 31;

    const float* ge  = g  + (size_t)e * TPE  * FDIM;
    const float* w2e = w2 + (size_t)e * FDIM * HDIM + hblk;

    // A-tile copy slice (64 rows x 32 cols, 8 f4/row)
    const int arow[4] = { (tid + 0*128) >> 3, (tid + 1*128) >> 3,
                          (tid + 2*128) >> 3, (tid + 3*128) >> 3 };
    const int acol    = (tid & 7) * 4;
    // B-tile copy slice (32 rows x 64 cols, 16 f4/row)
    const int brow[4] = { (tid + 0*128) >> 4, (tid + 1*128) >> 4,
                          (tid + 2*128) >> 4, (tid + 3*128) >> 4 };
    const int bcol    = (tid & 15) * 4;

    const v8f vzero = {0.f,0.f,0.f,0.f,0.f,0.f,0.f,0.f};
    v8f acc[4];
#pragma unroll
    for (int i = 0; i < 4; ++i) acc[i] = vzero;

    const int mrow = lane & 15;
    const int koff = (lane >> 4) << 1;
    const int ncol = lane & 15;

    // issue tile 0
#pragma unroll
    for (int i = 0; i < 4; ++i) {
        async_copy_b128(lds_off(&gs[0][arow[i] * AST + acol]),
                        ge + (size_t)arow[i] * FDIM + acol);
        async_copy_b128(lds_off(&bs[0][brow[i] * BST + bcol]),
                        w2e + (size_t)brow[i] * HDIM + bcol);
    }

    const int nTiles = FDIM / KC;
    for (int t = 0; t < nTiles; ++t) {
        wait_async0();
        __syncthreads();

        if (t + 1 < nTiles) {
            const int b  = (t + 1) & 1;
            const int k0 = (t + 1) * KC;
#pragma unroll
            for (int i = 0; i < 4; ++i) {
                async_copy_b128(lds_off(&gs[b][arow[i] * AST + acol]),
                                ge + (size_t)arow[i] * FDIM + k0 + acol);
                async_copy_b128(lds_off(&bs[b][brow[i] * BST + bcol]),
                                w2e + (size_t)(k0 + brow[i]) * HDIM + bcol);
            }
        }

        const float* ab = gs[t & 1];
        const float* bb = bs[t & 1];

#pragma unroll
        for (int kk = 0; kk < KC; kk += 4) {
            const int kb = kk + koff;
            v2f b;
            b.x = bb[(kb)     * BST + wave * 16 + ncol];
            b.y = bb[(kb + 1) * BST + wave * 16 + ncol];
#pragma unroll
            for (int mi = 0; mi < 4; ++mi) {
                v2f a;
                a.x = ab[(mi * 16 + mrow) * AST + kb];
                a.y = ab[(mi * 16 + mrow) * AST + kb + 1];
                acc[mi] = __builtin_amdgcn_wmma_f32_16x16x4_f32(
                    false, a, false, b, (short)0, acc[mi], false, false);
            }
        }
    }

    float* oe = out + (size_t)e * TPE * HDIM;
    const int nOut = hblk + wave * 16 + (lane & 15);
    const int mhi  = (lane >> 4) * 8;
#pragma unroll
    for (int mi = 0; mi < 4; ++mi) {
#pragma unroll
        for (int r = 0; r < 8; ++r) {
            int tt = mi * 16 + mhi + r;
            oe[(size_t)tt * HDIM + nOut] = acc[mi][r];
        }
    }
}

extern "C" void kernel_launch(void* const* d_in, const int* in_sizes, int n_in,
                              void* d_out, int out_size, void* d_ws, size_t ws_size,
                              hipStream_t stream) {
    const float* x  = (const float*)d_in[0];
    const float* w1 = (const float*)d_in[1];
    const float* v1 = (const float*)d_in[2];
    const float* w2 = (const float*)d_in[3];
    float* out = (float*)d_out;
    float* gws = (float*)d_ws;   // needs NEXP*TPE*FDIM*4 = 16 MiB scratch

    dim3 block(128);
    dim3 grid1(NEXP * (FDIM / 64));
    dim3 grid2(NEXP * (HDIM / 64));
    moe_glu_pass1<<<grid1, block, 0, stream>>>(x, w1, v1, gws);
    moe_glu_pass2<<<grid2, block, 0, stream>>>(gws, w2, out);
    (void)in_sizes; (void)n_in; (void)out_size; (void)ws_size;
}